// SpectralGCN_75909251989908
// MI455X (gfx1250) — compile-verified
//
#include <hip/hip_runtime.h>
#include <math.h>

#define N_NODES 50000
#define N_EDGES 800000
#define FEATS   128
#define BN_EPS  1e-5f

typedef __attribute__((ext_vector_type(2))) float v2f;
typedef __attribute__((ext_vector_type(8))) float v8f;

// ---------------------------------------------------------------------------
// P1: circulant spectral kernel s[d] (folds rfft * w -> irfft) + fused bias
// s[d] = (1/128)[a0 + (-1)^d a64 + 2*sum_{k=1..63}(a_k cos(pi k d/64) - b_k sin(pi k d/64))]
// bias[j] = b_fuse[j] + alpha * sum_k s[(j-k)&127] * b_in[k]
// ---------------------------------------------------------------------------
__global__ __launch_bounds__(128) void prep_s_bias(
    const float* __restrict__ cw, const float* __restrict__ b_fuse,
    const float* __restrict__ b_in, const float* __restrict__ alpha_p,
    float* __restrict__ s_out, float* __restrict__ bias_out) {
  __shared__ float ls[128];
  const int d = threadIdx.x;
  float acc = cw[0] + ((d & 1) ? -cw[128] : cw[128]);   // a0 + (-1)^d * a64
  for (int k = 1; k < 64; ++k) {
    const float ph = (float)(k * d) * (1.0f / 64.0f);
    acc += 2.0f * (cw[2 * k] * cospif(ph) - cw[2 * k + 1] * sinpif(ph));
  }
  const float sd = acc * (1.0f / 128.0f);
  s_out[d] = sd;
  ls[d] = sd;
  __syncthreads();
  const float alpha = alpha_p[0];
  float bb = 0.0f;
  for (int k = 0; k < 128; ++k) bb += ls[(d - k) & 127] * b_in[k];
  bias_out[d] = b_fuse[d] + alpha * bb;
}

// ---------------------------------------------------------------------------
// P2: build A = Wf1^T + alpha*W_in^T S^T and B = Wf2^T directly in
// WMMA B-fragment-major layout so the GEMM inner loop does one coalesced
// b64 load per fragment:
//   element (k=i, outcol=j) lives at
//   frag[(j>>4)*2048 + (i>>2)*64 + (((i>>1)&1)*16 + (j&15))*2 + (i&1)]
// (colTile-major, then K-step, then lane = half*16 + col, then pair elem)
// ---------------------------------------------------------------------------
__global__ __launch_bounds__(128) void prep_AB(
    const float* __restrict__ W_fuse, const float* __restrict__ W_in,
    const float* __restrict__ s, const float* __restrict__ alpha_p,
    float* __restrict__ Afrag, float* __restrict__ Bfrag) {
  __shared__ float ls[128];
  const int i = threadIdx.x;   // input feature (K)
  const int j = blockIdx.x;    // output feature (N)
  ls[i] = s[i];
  __syncthreads();
  float acc = 0.0f;
  for (int k = 0; k < 128; ++k) acc += W_in[k * 128 + i] * ls[(j - k) & 127];
  const int idx = (j >> 4) * 2048 + (i >> 2) * 64 +
                  (((i >> 1) & 1) * 16 + (j & 15)) * 2 + (i & 1);
  Afrag[idx] = W_fuse[j * 256 + i] + alpha_p[0] * acc;
  Bfrag[idx] = W_fuse[j * 256 + 128 + i];
}

// ---------------------------------------------------------------------------
// Edge degree count (float so the mean divide matches reference semantics)
// ---------------------------------------------------------------------------
__global__ __launch_bounds__(256) void edge_count(
    const long long* __restrict__ ei, float* __restrict__ cnt) {
  const int e = blockIdx.x * 256 + threadIdx.x;
  if (e >= N_EDGES) return;
  atomicAdd(&cnt[(int)ei[N_EDGES + e]], 1.0f);
}

// ---------------------------------------------------------------------------
// Scatter-add neighbor features: thread = (edge, 4-feature chunk)
// ---------------------------------------------------------------------------
__global__ __launch_bounds__(256) void edge_scatter(
    const long long* __restrict__ ei, const float* __restrict__ x,
    float* __restrict__ agg) {
  const unsigned tid = blockIdx.x * 256u + threadIdx.x;
  const unsigned e = tid >> 5;
  const unsigned f = (tid & 31u) * 4u;
  const int s = (int)ei[e];
  const int d = (int)ei[N_EDGES + e];
  const float4 v = *(const float4*)(x + (size_t)s * FEATS + f);
  float* p = agg + (size_t)d * FEATS + f;
  atomicAdd(p + 0, v.x);
  atomicAdd(p + 1, v.y);
  atomicAdd(p + 2, v.z);
  atomicAdd(p + 3, v.w);
}

// ---------------------------------------------------------------------------
// aggs = agg_sum / max(cnt, 1)
// ---------------------------------------------------------------------------
__global__ __launch_bounds__(256) void agg_norm(
    float* __restrict__ agg, const float* __restrict__ cnt) {
  const int tid = blockIdx.x * 256 + threadIdx.x;
  const float c = cnt[tid >> 7];
  agg[tid] *= 1.0f / fmaxf(c, 1.0f);
}

// ---------------------------------------------------------------------------
// Fused dual GEMM via V_WMMA_F32_16X16X4_F32:
//   out_pre = x @ A + aggs @ B + bias
// Block = 16 rows x 128 cols; 8 wave32s, one 16x16 C tile per wave.
// x/agg tiles staged straight into LDS with GLOBAL_LOAD_ASYNC_TO_LDS_B128
// (ASYNCcnt path, no VGPR round-trip); LDS stride 132 keeps rows 16B-aligned
// and fragment reads bank-conflict-free. B-fragments of A/B stream from L2
// as fully coalesced b64 loads thanks to the fragment-major layout.
// ---------------------------------------------------------------------------
__global__ __launch_bounds__(256) void fused_gemm(
    const float* __restrict__ x, const float* __restrict__ agg,
    const float* __restrict__ Afrag, const float* __restrict__ Bfrag,
    const float* __restrict__ bias, float* __restrict__ out) {
  __shared__ float lX[16 * 132];
  __shared__ float lG[16 * 132];
  const int row0 = blockIdx.x * 16;
  const int t = threadIdx.x;
  {
    const int r = t >> 5;           // 0..7 -> rows r and r+8
    const int c = (t & 31) * 4;     // 0..124, float4-granular
    const unsigned dX0 = (unsigned)(uintptr_t)&lX[r * 132 + c];
    const unsigned dX8 = (unsigned)(uintptr_t)&lX[(r + 8) * 132 + c];
    const unsigned dG0 = (unsigned)(uintptr_t)&lG[r * 132 + c];
    const unsigned dG8 = (unsigned)(uintptr_t)&lG[(r + 8) * 132 + c];
    const float* gX0 = &x[(size_t)(row0 + r) * FEATS + c];
    const float* gX8 = &x[(size_t)(row0 + r + 8) * FEATS + c];
    const float* gG0 = &agg[(size_t)(row0 + r) * FEATS + c];
    const float* gG8 = &agg[(size_t)(row0 + r + 8) * FEATS + c];
    asm volatile(
        "global_load_async_to_lds_b128 %0, %4, off\n\t"
        "global_load_async_to_lds_b128 %1, %5, off\n\t"
        "global_load_async_to_lds_b128 %2, %6, off\n\t"
        "global_load_async_to_lds_b128 %3, %7, off\n\t"
        "s_wait_asynccnt 0"
        :
        : "v"(dX0), "v"(dX8), "v"(dG0), "v"(dG8),
          "v"(gX0), "v"(gX8), "v"(gG0), "v"(gG8)
        : "memory");
  }
  __syncthreads();

  const int lane = t & 31;
  const int col0 = (t >> 5) * 16;          // wave -> 16-col tile
  const int m    = lane & 15;              // A: row, B: col
  const int kh   = (lane >> 4) * 2;        // K sub-offset per lane half
  const float* Ap = Afrag + col0 * 128 + lane * 2;   // colTile*2048 + lane*2
  const float* Bp = Bfrag + col0 * 128 + lane * 2;
  v8f acc = {};

#pragma unroll 4
  for (int kk = 0; kk < 128; kk += 4) {
    const int k = kk + kh;
    const int fo = (kk >> 2) * 64;
    v2f a;
    a.x = lX[m * 132 + k];
    a.y = lX[m * 132 + k + 1];
    const v2f b = *(const v2f*)(Ap + fo);
    acc = __builtin_amdgcn_wmma_f32_16x16x4_f32(false, a, false, b,
                                                (short)0, acc, false, false);
    v2f a2;
    a2.x = lG[m * 132 + k];
    a2.y = lG[m * 132 + k + 1];
    const v2f b2 = *(const v2f*)(Bp + fo);
    acc = __builtin_amdgcn_wmma_f32_16x16x4_f32(false, a2, false, b2,
                                                (short)0, acc, false, false);
  }

  // C layout: VGPR i -> M = i (lanes 0-15), M = i+8 (lanes 16-31); N = lane&15
  const int col   = col0 + m;
  const float bc  = bias[col];
  const int rbase = row0 + (lane >> 4) * 8;
#pragma unroll
  for (int i = 0; i < 8; ++i)
    out[(size_t)(rbase + i) * FEATS + col] = acc[i] + bc;
}

// ---------------------------------------------------------------------------
// BatchNorm stats: per-feature sum / sumsq over all rows
// ---------------------------------------------------------------------------
__global__ __launch_bounds__(256) void bn_stats(
    const float* __restrict__ pre, float* __restrict__ gsum,
    float* __restrict__ gsq) {
  __shared__ float ss[256], sq[256];
  const int t = threadIdx.x;
  const int f = t & 127;
  const int half = t >> 7;
  float s = 0.0f, q = 0.0f;
  for (int n = blockIdx.x * 2 + half; n < N_NODES; n += gridDim.x * 2) {
    const float v = pre[(size_t)n * FEATS + f];
    s += v;
    q += v * v;
  }
  ss[t] = s;
  sq[t] = q;
  __syncthreads();
  if (t < 128) {
    atomicAdd(&gsum[f], ss[t] + ss[t + 128]);
    atomicAdd(&gsq[f], sq[t] + sq[t + 128]);
  }
}

// ---------------------------------------------------------------------------
// In-place BN (training stats, biased var) + exact GELU
// ---------------------------------------------------------------------------
__global__ __launch_bounds__(256) void bn_gelu(
    float* __restrict__ out, const float* __restrict__ gsum,
    const float* __restrict__ gsq, const float* __restrict__ gamma,
    const float* __restrict__ beta) {
  const int tid = blockIdx.x * 256 + threadIdx.x;
  const int f = tid & 127;
  const float mean = gsum[f] * (1.0f / N_NODES);
  const float var = gsq[f] * (1.0f / N_NODES) - mean * mean;
  float v = (out[tid] - mean) * rsqrtf(var + BN_EPS) * gamma[f] + beta[f];
  out[tid] = 0.5f * v * (1.0f + erff(v * 0.70710678118654752f));
}

// ---------------------------------------------------------------------------
extern "C" void kernel_launch(void* const* d_in, const int* in_sizes, int n_in,
                              void* d_out, int out_size, void* d_ws,
                              size_t ws_size, hipStream_t stream) {
  const float*     x      = (const float*)d_in[0];
  const long long* ei     = (const long long*)d_in[1];   // int64 edge_index
  const float*     W_fuse = (const float*)d_in[2];
  const float*     b_fuse = (const float*)d_in[3];
  const float*     W_in   = (const float*)d_in[4];
  const float*     b_in   = (const float*)d_in[5];
  const float*     cw     = (const float*)d_in[6];
  const float*     alpha  = (const float*)d_in[7];
  const float*     gamma  = (const float*)d_in[8];
  const float*     beta   = (const float*)d_in[9];
  float* out = (float*)d_out;

  char* ws = (char*)d_ws;
  // layout: [agg 25.6MB][cnt 200KB][gsum 512][gsq 512] <- zeroed each call
  //         [Afrag 64KB][Bfrag 64KB][s 512][bias 512]
  float* agg  = (float*)(ws + 0);
  float* cnt  = (float*)(ws + 25600000);
  float* gsum = (float*)(ws + 25800000);
  float* gsq  = (float*)(ws + 25800512);
  float* A    = (float*)(ws + 25801024);
  float* B    = (float*)(ws + 25866560);
  float* sbuf = (float*)(ws + 25932096);
  float* bias = (float*)(ws + 25932608);

  hipMemsetAsync(ws, 0, 25801024, stream);  // agg + cnt + gsum + gsq

  prep_s_bias<<<1, 128, 0, stream>>>(cw, b_fuse, b_in, alpha, sbuf, bias);
  prep_AB<<<128, 128, 0, stream>>>(W_fuse, W_in, sbuf, alpha, A, B);
  edge_count<<<(N_EDGES + 255) / 256, 256, 0, stream>>>(ei, cnt);
  edge_scatter<<<(N_EDGES * 32) / 256, 256, 0, stream>>>(ei, x, agg);
  agg_norm<<<(N_NODES * FEATS) / 256, 256, 0, stream>>>(agg, cnt);
  fused_gemm<<<N_NODES / 16, 256, 0, stream>>>(x, agg, A, B, bias, out);
  bn_stats<<<250, 256, 0, stream>>>(out, gsum, gsq);
  bn_gelu<<<(N_NODES * FEATS) / 256, 256, 0, stream>>>(out, gsum, gsq, gamma, beta);
}